// TripletSNDCGLoss_32530082300105
// MI455X (gfx1250) — compile-verified
//
#include <hip/hip_runtime.h>
#include <hip/hip_bf16.h>
#include <math.h>

#define B_SZ   384
#define D_SZ   1024
#define DT_SZ  300
#define MARGIN 0.2f
#define TAU    0.01f
#define NDCG_W 0.5f

typedef __attribute__((ext_vector_type(2))) float v2f;
typedef __attribute__((ext_vector_type(8))) float v8f;

// ---------------------------------------------------------------------------
// Block reductions for blockDim.x == 384 (pad shared buffer to 512).
// ---------------------------------------------------------------------------
__device__ __forceinline__ float block_reduce_sum384(float x, float* red) {
    const int tid = threadIdx.x;
    red[tid] = x;
    if (tid < 128) red[tid + 384] = 0.0f;
    __syncthreads();
    #pragma unroll
    for (int s = 256; s > 0; s >>= 1) {
        if (tid < s) red[tid] += red[tid + s];
        __syncthreads();
    }
    float r = red[0];
    __syncthreads();
    return r;
}

__device__ __forceinline__ float block_reduce_max384(float x, float* red) {
    const int tid = threadIdx.x;
    red[tid] = x;
    if (tid < 128) red[tid + 384] = -1e30f;
    __syncthreads();
    #pragma unroll
    for (int s = 256; s > 0; s >>= 1) {
        if (tid < s) red[tid] = fmaxf(red[tid], red[tid + s]);
        __syncthreads();
    }
    float r = red[0];
    __syncthreads();
    return r;
}

// ---------------------------------------------------------------------------
// 1) L2-normalize text embedding rows.
//    Rows 0..5*B-1  : v_text_emb[b][cap][:]  -> vtn[cap][b][:]  ([5,B,DT])
//    Rows 5*B..6*B-1: t_text_emb[b][:]       -> ttn[b][:]
//    One wave (32 lanes) per row; 8 waves / 256-thread block.
// ---------------------------------------------------------------------------
__global__ __launch_bounds__(256)
void normalize_rows_kernel(const float* __restrict__ v_text,
                           const float* __restrict__ t_text,
                           float* __restrict__ vtn,
                           float* __restrict__ ttn) {
    const int wave = (int)((blockIdx.x * blockDim.x + threadIdx.x) >> 5);
    const int lane = (int)(threadIdx.x & 31);
    const int nrows = 6 * B_SZ;          // 2304
    if (wave >= nrows) return;

    const float* src;
    float* dst;
    if (wave < 5 * B_SZ) {
        const int cap = wave / B_SZ;
        const int b   = wave % B_SZ;
        src = v_text + ((size_t)b * 5 + cap) * DT_SZ;
        dst = vtn    + ((size_t)cap * B_SZ + b) * DT_SZ;
    } else {
        const int b = wave - 5 * B_SZ;
        src = t_text + (size_t)b * DT_SZ;
        dst = ttn    + (size_t)b * DT_SZ;
    }

    float ss = 0.0f;
    for (int i = lane; i < DT_SZ; i += 32) {
        float x = src[i];
        ss += x * x;
    }
    #pragma unroll
    for (int off = 16; off > 0; off >>= 1)
        ss += __shfl_xor(ss, off, 32);
    const float inv = 1.0f / sqrtf(ss);
    for (int i = lane; i < DT_SZ; i += 32)
        dst[i] = src[i] * inv;
}

// ---------------------------------------------------------------------------
// 2) scores = v @ t.T via V_WMMA_F32_16X16X4_F32. One wave per 16x16 tile.
//    A fragment (16x4 f32): lane L holds A[L%16][k0], A[L%16][k0+1], k0=kk+2*(L/16)
//    B fragment (4x16 f32): lane L holds B[k0][L%16], B[k0+1][L%16]
//      with B[k][n] = t[n][k]  => same contiguous row-pair load pattern as A.
//    D (16x16 f32): VGPR g, lane L -> row g + 8*(L/16), col L%16.
// ---------------------------------------------------------------------------
__global__ __launch_bounds__(32)
void scores_wmma_kernel(const float* __restrict__ v,
                        const float* __restrict__ t,
                        float* __restrict__ scores) {
    const int ti   = blockIdx.y;
    const int tj   = blockIdx.x;
    const int lane = (int)threadIdx.x;
    const int half = lane >> 4;
    const int p    = lane & 15;

    const float* arow = v + (size_t)(ti * 16 + p) * D_SZ;
    const float* brow = t + (size_t)(tj * 16 + p) * D_SZ;

    v8f c = {0.f, 0.f, 0.f, 0.f, 0.f, 0.f, 0.f, 0.f};
    #pragma unroll 8
    for (int kk = 0; kk < D_SZ; kk += 4) {
        v2f a = *(const v2f*)(arow + kk + 2 * half);
        v2f b = *(const v2f*)(brow + kk + 2 * half);
        c = __builtin_amdgcn_wmma_f32_16x16x4_f32(false, a, false, b,
                                                  (short)0, c, false, false);
    }

    #pragma unroll
    for (int g = 0; g < 8; ++g) {
        const int i = ti * 16 + g + 8 * half;
        scores[(size_t)i * B_SZ + tj * 16 + p] = c[g];
    }
}

// ---------------------------------------------------------------------------
// 3) rel[i][j] = (1 + max_cap dot(vtn[cap][i], ttn[j])) / 2, diag forced to 1.
//    gain = 2^rel - 1. One wave per 16x16 tile; WMMA over K=300, 5 captions.
// ---------------------------------------------------------------------------
__global__ __launch_bounds__(32)
void rel_wmma_kernel(const float* __restrict__ vtn,
                     const float* __restrict__ ttn,
                     float* __restrict__ rel,
                     float* __restrict__ gain) {
    const int ti   = blockIdx.y;
    const int tj   = blockIdx.x;
    const int lane = (int)threadIdx.x;
    const int half = lane >> 4;
    const int p    = lane & 15;

    const float* brow = ttn + (size_t)(tj * 16 + p) * DT_SZ;

    float m[8];
    #pragma unroll
    for (int g = 0; g < 8; ++g) m[g] = -1e30f;

    for (int cap = 0; cap < 5; ++cap) {
        const float* arow = vtn + ((size_t)cap * B_SZ + (ti * 16 + p)) * DT_SZ;
        v8f c = {0.f, 0.f, 0.f, 0.f, 0.f, 0.f, 0.f, 0.f};
        #pragma unroll 5
        for (int kk = 0; kk < DT_SZ; kk += 4) {
            v2f a = *(const v2f*)(arow + kk + 2 * half);
            v2f b = *(const v2f*)(brow + kk + 2 * half);
            c = __builtin_amdgcn_wmma_f32_16x16x4_f32(false, a, false, b,
                                                      (short)0, c, false, false);
        }
        #pragma unroll
        for (int g = 0; g < 8; ++g) m[g] = fmaxf(m[g], c[g]);
    }

    const int j = tj * 16 + p;
    #pragma unroll
    for (int g = 0; g < 8; ++g) {
        const int i = ti * 16 + g + 8 * half;
        float r = 0.5f * (1.0f + m[g]);
        if (i == j) r = 1.0f;
        rel[(size_t)i * B_SZ + j]  = r;
        gain[(size_t)i * B_SZ + j] = exp2f(r) - 1.0f;
    }
}

// ---------------------------------------------------------------------------
// 4) Pairwise triplet maxes: per index b,
//    ltmax[b] = max_j!=b clip(scores[b][j] - scores[b][b] + M, 0)
//    lvmax[b] = max_i!=b clip(scores[i][b] - scores[b][b] + M, 0)
// ---------------------------------------------------------------------------
__global__ __launch_bounds__(384)
void pairwise_max_kernel(const float* __restrict__ scores,
                         float* __restrict__ ltmax,
                         float* __restrict__ lvmax) {
    __shared__ float red[512];
    const int b   = blockIdx.x;
    const int tid = (int)threadIdx.x;
    const float diag = scores[(size_t)b * B_SZ + b];

    float x = fmaxf(scores[(size_t)b * B_SZ + tid] - diag + MARGIN, 0.0f);
    if (tid == b) x = 0.0f;
    const float rmax = block_reduce_max384(x, red);

    float y = fmaxf(scores[(size_t)tid * B_SZ + b] - diag + MARGIN, 0.0f);
    if (tid == b) y = 0.0f;
    const float cmax = block_reduce_max384(y, red);

    if (tid == 0) {
        ltmax[b] = rmax;
        lvmax[b] = cmax;
    }
}

// ---------------------------------------------------------------------------
// 5) Per-row rank + NDCG (the O(B^3) part). Block i, thread j.
//    soft_rk[j] = 1 + sum_{k!=j} 1/(1+exp(clip((s[j]-s[k])/TAU, -50, 50)))
//    hard_rk[j] = 1 + #{k : s[k] > s[j]}
// ---------------------------------------------------------------------------
__device__ __forceinline__ void rank_terms(const float* __restrict__ srow,
                                           const float* __restrict__ grow,
                                           int j, float* dcg_t, float* rdcg_t) {
    const float sj = srow[j];
    float soft = 1.0f;
    int   hard = 1;
    for (int k = 0; k < B_SZ; ++k) {
        const float sk = srow[k];
        float e = (sj - sk) * (1.0f / TAU);     // == clip(-diff/tau, ...)
        e = fminf(fmaxf(e, -50.0f), 50.0f);
        const float sig = 1.0f / (1.0f + exp2f(e * 1.4426950408889634f));
        soft += (k != j) ? sig : 0.0f;
        hard += (sk > sj) ? 1 : 0;
    }
    *dcg_t  = grow[j] / log2f(1.0f + soft);
    *rdcg_t = grow[j] / log2f(1.0f + (float)hard);
}

__global__ __launch_bounds__(384)
void rank_ndcg_kernel(const float* __restrict__ scores,
                      const float* __restrict__ rel,
                      const float* __restrict__ gain,
                      float* __restrict__ ndcg_t, float* __restrict__ rndcg_t,
                      float* __restrict__ ndcg_v, float* __restrict__ rndcg_v) {
    __shared__ float srow[B_SZ];
    __shared__ float grow[B_SZ];
    __shared__ float rrow[B_SZ];
    __shared__ float red[512];

    const int i = blockIdx.x;
    const int j = (int)threadIdx.x;

    grow[j] = gain[(size_t)i * B_SZ + j];
    rrow[j] = rel[(size_t)i * B_SZ + j];
    srow[j] = scores[(size_t)i * B_SZ + j];
    __syncthreads();

    // IDCG from hard ranks over relevance row
    {
        const float rj = rrow[j];
        int cnt = 1;
        for (int k = 0; k < B_SZ; ++k)
            cnt += (rrow[k] > rj) ? 1 : 0;
        const float term = grow[j] / log2f(1.0f + (float)cnt);
        const float idcg = block_reduce_sum384(term, red);

        // t-direction: row i of scores
        float dt_, rt_;
        rank_terms(srow, grow, j, &dt_, &rt_);
        const float dcg  = block_reduce_sum384(dt_, red);
        const float rdcg = block_reduce_sum384(rt_, red);
        if (j == 0) {
            ndcg_t[i]  = dcg / idcg;
            rndcg_t[i] = rdcg / idcg;
        }

        // v-direction: column i of scores
        __syncthreads();
        srow[j] = scores[(size_t)j * B_SZ + i];
        __syncthreads();
        float dv_, rv_;
        rank_terms(srow, grow, j, &dv_, &rv_);
        const float dcgv  = block_reduce_sum384(dv_, red);
        const float rdcgv = block_reduce_sum384(rv_, red);
        if (j == 0) {
            ndcg_v[i]  = dcgv / idcg;
            rndcg_v[i] = rdcgv / idcg;
        }
    }
}

// ---------------------------------------------------------------------------
// 6) Finalize: means + loss composition -> 7 outputs.
// ---------------------------------------------------------------------------
__global__ __launch_bounds__(384)
void finalize_kernel(const float* __restrict__ ltm, const float* __restrict__ lvm,
                     const float* __restrict__ ndt, const float* __restrict__ rdt,
                     const float* __restrict__ ndv, const float* __restrict__ rdv,
                     float* __restrict__ out) {
    __shared__ float red[512];
    const int j = (int)threadIdx.x;
    const float s_lt = block_reduce_sum384(ltm[j], red);
    const float s_lv = block_reduce_sum384(lvm[j], red);
    const float s_nt = block_reduce_sum384(ndt[j], red);
    const float s_rt = block_reduce_sum384(rdt[j], red);
    const float s_nv = block_reduce_sum384(ndv[j], red);
    const float s_rv = block_reduce_sum384(rdv[j], red);
    if (j == 0) {
        const float invB = 1.0f / (float)B_SZ;
        const float pairwise = 0.5f * (s_lt * invB + s_lv * invB);
        const float m_nt = s_nt * invB, m_rt = s_rt * invB;
        const float m_nv = s_nv * invB, m_rv = s_rv * invB;
        const float listwise = 0.5f * ((1.0f - m_nt) + (1.0f - m_nv));
        const float loss = (1.0f - NDCG_W) * pairwise + NDCG_W * listwise;
        out[0] = loss;
        out[1] = pairwise;
        out[2] = listwise;
        out[3] = m_nt;
        out[4] = m_rt;
        out[5] = m_nv;
        out[6] = m_rv;
    }
}

// ---------------------------------------------------------------------------
// Launcher
// ---------------------------------------------------------------------------
extern "C" void kernel_launch(void* const* d_in, const int* in_sizes, int n_in,
                              void* d_out, int out_size, void* d_ws, size_t ws_size,
                              hipStream_t stream) {
    (void)in_sizes; (void)n_in; (void)out_size; (void)ws_size;

    const float* v   = (const float*)d_in[0];   // [384,1024]
    const float* t   = (const float*)d_in[1];   // [384,1024]
    const float* vte = (const float*)d_in[2];   // [384,5,300]
    const float* tte = (const float*)d_in[3];   // [384,300]
    float* out = (float*)d_out;                 // 7 floats

    float* ws      = (float*)d_ws;
    float* scores  = ws;                        // 384*384 = 147456
    float* rel     = scores + 147456;           // 147456
    float* gain    = rel    + 147456;           // 147456
    float* vtn     = gain   + 147456;           // 5*384*300 = 576000
    float* ttn     = vtn    + 576000;           // 384*300   = 115200
    float* ltm     = ttn    + 115200;           // 384
    float* lvm     = ltm    + 384;              // 384
    float* ndt     = lvm    + 384;              // 384
    float* rdt     = ndt    + 384;              // 384
    float* ndv     = rdt    + 384;              // 384
    float* rdv     = ndv    + 384;              // 384
    // total: 1,135,872 floats ~ 4.34 MB

    normalize_rows_kernel<<<288, 256, 0, stream>>>(vte, tte, vtn, ttn);

    dim3 tiles(B_SZ / 16, B_SZ / 16);           // 24 x 24
    scores_wmma_kernel<<<tiles, 32, 0, stream>>>(v, t, scores);
    rel_wmma_kernel<<<tiles, 32, 0, stream>>>(vtn, ttn, rel, gain);

    pairwise_max_kernel<<<B_SZ, 384, 0, stream>>>(scores, ltm, lvm);
    rank_ndcg_kernel<<<B_SZ, 384, 0, stream>>>(scores, rel, gain, ndt, rdt, ndv, rdv);
    finalize_kernel<<<1, 384, 0, stream>>>(ltm, lvm, ndt, rdt, ndv, rdv, out);
}